// Quantizer_49615462204013
// MI455X (gfx1250) — compile-verified
//
#include <hip/hip_runtime.h>

// Per-group (128 fp32) power-of-two int8 quant + dequant (fused path).
// One group = one wave32 * float4/lane. Bandwidth-bound:
// 256MiB in + 256MiB out -> ~23us floor @ 23.3 TB/s.
//
// Round-2 improvements:
//  * readfirstlane the wave index -> wave-uniform scalar guards, saddr-form
//    global loads, and a guard-free fast path so all 8 NT b128 loads clause.
//  * ds_swizzle_b32 immediate-xor butterfly (no shuffle index VALU setup).
//  * 8 groups in flight per wave for MLP + interleaved reduction chains.

typedef float v4f __attribute__((ext_vector_type(4)));

#define GPW 8   // groups per wave (8 x global_load_b128 in flight)
#define WPB 8   // wave32s per 256-thread block

// One butterfly step on all GPW chains via ds_swizzle_b32 with immediate
// pattern (offset[15]=0: group-of-32; bits[14:10]=xor, [4:0]=and=0x1f).
template <int IMM>
__device__ __forceinline__ void swz_max_step(float (&m)[GPW]) {
#pragma unroll
  for (int k = 0; k < GPW; ++k) {
    int s = __builtin_amdgcn_ds_swizzle(__builtin_bit_cast(int, m[k]), IMM);
    m[k] = fmaxf(m[k], __builtin_bit_cast(float, s));
  }
}

// PoT quantize+dequantize one float4 given the group abs-max.
__device__ __forceinline__ v4f pot_qdq(v4f v, float absmax, int smax) {
  // scale0 = absmax / 127 (true divide, matches reference)
  const float sc0 = absmax / 127.0f;
  // ceil(log2(sc0)) exactly from fp32 bits: e + (mantissa != 0).
  // sc0==0 -> e=-127 -> clamps to 0 -> scale=1 (matches reference).
  const unsigned b = __builtin_bit_cast(unsigned, sc0);
  int e = (int)((b >> 23) & 0xFFu) - 127 + ((b & 0x7FFFFFu) ? 1 : 0);
  e = min(max(e, 0), smax);
  // 2^e with e in [0, 2^W-1] is exact in fp16 -> reference's fp16 cast is
  // identity; 2^-e is exact -> multiply == divide.
  const float scale = __builtin_bit_cast(float, (unsigned)(e + 127) << 23);
  const float inv   = __builtin_bit_cast(float, (unsigned)(127 - e) << 23);
  v4f q;
  q.x = fminf(fmaxf(__builtin_rintf(v.x * inv), -128.0f), 127.0f) * scale;
  q.y = fminf(fmaxf(__builtin_rintf(v.y * inv), -128.0f), 127.0f) * scale;
  q.z = fminf(fmaxf(__builtin_rintf(v.z * inv), -128.0f), 127.0f) * scale;
  q.w = fminf(fmaxf(__builtin_rintf(v.w * inv), -128.0f), 127.0f) * scale;
  return q;
}

__device__ __forceinline__ float abs_max4(v4f v) {
  return fmaxf(fmaxf(__builtin_fabsf(v.x), __builtin_fabsf(v.y)),
               fmaxf(__builtin_fabsf(v.z), __builtin_fabsf(v.w)));
}

__global__ __launch_bounds__(256) void quant_pot_kernel(
    const float* __restrict__ in, float* __restrict__ out,
    const int* __restrict__ wbits_p, int n_groups) {

  const int lane = threadIdx.x & 31;
  // Force the wave index into an SGPR: guards become scalar branches and
  // global addresses become SGPR-base + lane-offset (saddr form).
  const int wib  = __builtin_amdgcn_readfirstlane(threadIdx.x >> 5);
  const int g0   = (blockIdx.x * WPB + wib) * GPW;

  const int smax = (1 << *wbits_p) - 1;  // uniform scalar load

  const v4f* pin  = (const v4f*)in + (size_t)g0 * 32 + lane;
  v4f*       pout = (v4f*)out      + (size_t)g0 * 32 + lane;

  v4f   v[GPW];
  float m[GPW];

  if (g0 + GPW <= n_groups) {
    // ---- fast path: wave-uniform, full EXEC, no guards ----
    // 8 independent NT b128 loads issued back-to-back (clause-able).
#pragma unroll
    for (int k = 0; k < GPW; ++k)
      v[k] = __builtin_nontemporal_load(pin + k * 32);   // +k*512B imm offset
#pragma unroll
    for (int k = 0; k < GPW; ++k)
      m[k] = abs_max4(v[k]);

    // wave32 abs-max butterfly: 5 ds_swizzle steps x 8 interleaved chains.
    swz_max_step<0x401f>(m);  // xor 16
    swz_max_step<0x201f>(m);  // xor 8
    swz_max_step<0x101f>(m);  // xor 4
    swz_max_step<0x081f>(m);  // xor 2
    swz_max_step<0x041f>(m);  // xor 1

#pragma unroll
    for (int k = 0; k < GPW; ++k)
      __builtin_nontemporal_store(pot_qdq(v[k], m[k], smax), pout + k * 32);
  } else {
    // ---- tail path (never taken for 8192x8192; scalar-guarded) ----
#pragma unroll
    for (int k = 0; k < GPW; ++k) {
      if (g0 + k < n_groups) {
        v[k] = __builtin_nontemporal_load(pin + k * 32);
        m[k] = abs_max4(v[k]);
      } else {
        m[k] = 0.0f;
      }
    }
    swz_max_step<0x401f>(m);
    swz_max_step<0x201f>(m);
    swz_max_step<0x101f>(m);
    swz_max_step<0x081f>(m);
    swz_max_step<0x041f>(m);
#pragma unroll
    for (int k = 0; k < GPW; ++k) {
      if (g0 + k < n_groups)
        __builtin_nontemporal_store(pot_qdq(v[k], m[k], smax), pout + k * 32);
    }
  }
}

extern "C" void kernel_launch(void* const* d_in, const int* in_sizes, int n_in,
                              void* d_out, int out_size, void* d_ws, size_t ws_size,
                              hipStream_t stream) {
  const float* t     = (const float*)d_in[0];
  // d_in[1] = fused (always 1; reference implements only the fused path)
  const int*   wbits = (const int*)d_in[2];

  const int n        = in_sizes[0];              // 8192*8192
  const int n_groups = n / 128;                  // 524,288
  const int gpb      = GPW * WPB;                // 64 groups per block
  const int blocks   = (n_groups + gpb - 1) / gpb;   // 8192 blocks

  quant_pot_kernel<<<blocks, WPB * 32, 0, stream>>>(
      t, (float*)d_out, wbits, n_groups);
}